// ReconNet_60387240182489
// MI455X (gfx1250) — compile-verified
//
#include <hip/hip_runtime.h>
#include <math.h>

#define VOXEL_SIZE 0.04f
#define C_CH 24
#define H_IM 120
#define W_IM 160
#define V_NUM 9
#define OUT_CH 25
#define PV_STRIDE 31   // odd => conflict-free; holds comps 0..26 + 4 pad

typedef __attribute__((ext_vector_type(2))) float v2f;
typedef __attribute__((ext_vector_type(8))) float v8f;

// ---------------------------------------------------------------------------
// Kernel 1: transpose feats (V,C,H,W) -> (V,H,W,C) so bilinear corners are
// contiguous 96B rows (6 x float4).
// ---------------------------------------------------------------------------
__global__ void transpose_feats(const float* __restrict__ f,
                                float* __restrict__ ft, int total) {
    int i = blockIdx.x * blockDim.x + threadIdx.x;
    if (i >= total) return;
    int w = i % W_IM; int t = i / W_IM;
    int h = t % H_IM; t /= H_IM;
    int c = t % C_CH; int v = t / C_CH;
    ft[(((size_t)v * H_IM + h) * W_IM + w) * C_CH + c] = f[i];
}

// ---------------------------------------------------------------------------
// Kernel 2: main back-projection. Projection via V_WMMA_F32_16X16X4_F32:
//   D[r][n] = sum_k proj_row[r][k] * wh[k][n]
// A0 rows 0..14 = comps 0..14 (views 0..4), row 15 = comp 15 (dup of A1 row 0)
// A1 rows 0..11 = comps 15..26 (views 5..8), rows 12..15 = don't-care (padded)
// => every lane stores all 8 D elements of both results unconditionally.
// ---------------------------------------------------------------------------
__global__ __launch_bounds__(256) void backproject(
    const int*   __restrict__ coords,   // (N,4)
    const float* __restrict__ origin,   // (3)
    const float* __restrict__ ft,       // (V,H,W,C) transposed feats
    const float* __restrict__ proj,     // (V,4,4)
    float* __restrict__ out_feat,       // (N,25)
    float* __restrict__ out_cnt,        // (N)
    float* __restrict__ zbar_ws,        // (N)
    float* __restrict__ partials,       // (numBlocks,3)
    int N)
{
    __shared__ float pv[8 * 32 * PV_STRIDE];
    __shared__ float wred[8 * 3];

    const int tid  = threadIdx.x;
    const int wave = tid >> 5;
    const int lane = tid & 31;
    const int l15  = lane & 15;
    const bool hi  = lane >= 16;

    // ---- A matrices (projection rows), layout per ISA:
    // lane m:    VGPR0 = A[m][0], VGPR1 = A[m][1]
    // lane m+16: VGPR0 = A[m][2], VGPR1 = A[m][3]
    const int k0 = hi ? 2 : 0;
    v2f A0, A1;
    {
        const int m = l15;
        // A0: rows 0..14 -> (view m/3, comp m%3); row 15 -> (view 5, comp 0)
        int v0 = (m < 15) ? (m / 3) : 5;
        int c0 = (m < 15) ? (m % 3) : 0;
        const float* pr0 = proj + (size_t)(v0 * 4 + c0) * 4;
        A0.x = pr0[k0]; A0.y = pr0[k0 + 1];
        // A1: rows 0..11 -> (view 5+m/3, comp m%3); rows 12..15 don't-care
        int mm = (m < 12) ? m : 0;
        const float* pr1 = proj + (size_t)((5 + mm / 3) * 4 + (mm % 3)) * 4;
        A1.x = pr1[k0]; A1.y = pr1[k0 + 1];
    }

    const float ox = origin[0], oy = origin[1], oz = origin[2];
    const int base = blockIdx.x * 256 + wave * 32;

    // ---- two voxel groups of 16: build B (4x16), WMMA, scatter D to LDS
#pragma unroll
    for (int g = 0; g < 2; ++g) {
        int n  = base + g * 16 + l15;
        int nc = n < N ? n : N - 1;
        const int* cp = coords + (size_t)nc * 4;
        float wx = (float)cp[1] * VOXEL_SIZE + ox;
        float wy = (float)cp[2] * VOXEL_SIZE + oy;
        float wz = (float)cp[3] * VOXEL_SIZE + oz;
        // B layout: lane n: VGPR0=B[0][n]=wx, VGPR1=B[1][n]=wy
        //           lane n+16: VGPR0=B[2][n]=wz, VGPR1=B[3][n]=1
        v2f B;
        B.x = hi ? wz : wx;
        B.y = hi ? 1.0f : wy;
        v8f zero = {0.f, 0.f, 0.f, 0.f, 0.f, 0.f, 0.f, 0.f};
        v8f D0 = __builtin_amdgcn_wmma_f32_16x16x4_f32(
            false, A0, false, B, (short)0, zero, false, false);
        v8f D1 = __builtin_amdgcn_wmma_f32_16x16x4_f32(
            false, A1, false, B, (short)0, zero, false, false);

        // D element r in this lane = D[row = r + (hi?8:0)][col = l15]
        // D0 -> comps row (0..15); D1 -> comps 15+row (15..30, 27..30 padded)
        float* pvv = pv + (size_t)(wave * 32 + g * 16 + l15) * PV_STRIDE;
        const int rbase = hi ? 8 : 0;
#pragma unroll
        for (int r = 0; r < 8; ++r) {
            pvv[rbase + r]      = D0[r];   // unconditional
            pvv[15 + rbase + r] = D1[r];   // unconditional (index<=30<PV_STRIDE)
        }
    }
    __syncthreads();

    // ---- per-lane gather for voxel base+lane
    const int  n    = base + lane;
    const bool live = n < N;
    const float* mypv = pv + (size_t)(wave * 32 + lane) * PV_STRIDE;

    float acc[C_CH];
#pragma unroll
    for (int c = 0; c < C_CH; ++c) acc[c] = 0.f;
    float cnt = 0.f, zsum = 0.f;

    for (int v = 0; v < V_NUM; ++v) {
        float X = mypv[v * 3 + 0];
        float Y = mypv[v * 3 + 1];
        float Z = mypv[v * 3 + 2];
        float rz = 1.0f / Z;              // one divide instead of two
        float px = X * rz, py = Y * rz;
        bool inb = (Z > 0.f) & (px >= 0.f) & (px <= (float)(W_IM - 1)) &
                   (py >= 0.f) & (py <= (float)(H_IM - 1));
        if (!inb) continue;
        cnt  += 1.f;
        zsum += Z;

        float x0 = floorf(px), y0 = floorf(py);
        float wx1 = px - x0, wy1 = py - y0;
        float wx0 = 1.f - wx1, wy0 = 1.f - wy1;
        int xi0 = (int)x0, yi0 = (int)y0;
        int xi1 = (xi0 + 1 < W_IM) ? xi0 + 1 : W_IM - 1;  // clamped; weight=0 at edge
        int yi1 = (yi0 + 1 < H_IM) ? yi0 + 1 : H_IM - 1;

        const float* fv = ft + (size_t)v * H_IM * W_IM * C_CH;
        const float4* p00 = (const float4*)(fv + ((size_t)yi0 * W_IM + xi0) * C_CH);
        const float4* p01 = (const float4*)(fv + ((size_t)yi0 * W_IM + xi1) * C_CH);
        const float4* p10 = (const float4*)(fv + ((size_t)yi1 * W_IM + xi0) * C_CH);
        const float4* p11 = (const float4*)(fv + ((size_t)yi1 * W_IM + xi1) * C_CH);
        float w00 = wx0 * wy0, w01 = wx1 * wy0, w10 = wx0 * wy1, w11 = wx1 * wy1;
#pragma unroll
        for (int j = 0; j < C_CH / 4; ++j) {
            float4 a = p00[j], b = p01[j], c4 = p10[j], d4 = p11[j];
            acc[4 * j + 0] += w00 * a.x + w01 * b.x + w10 * c4.x + w11 * d4.x;
            acc[4 * j + 1] += w00 * a.y + w01 * b.y + w10 * c4.y + w11 * d4.y;
            acc[4 * j + 2] += w00 * a.z + w01 * b.z + w10 * c4.z + w11 * d4.z;
            acc[4 * j + 3] += w00 * a.w + w01 * b.w + w10 * c4.w + w11 * d4.w;
        }
    }

    const float denom = (cnt > 0.f) ? cnt : 1.f;
    const float zbar  = zsum / denom;
    if (live) {
        float* orow = out_feat + (size_t)n * OUT_CH;
#pragma unroll
        for (int c = 0; c < C_CH; ++c) orow[c] = acc[c] / denom;
        out_cnt[n] = cnt;
        zbar_ws[n] = zbar;
    }

    // ---- deterministic block partials: S0=sum valid, S1=sum zbar, S2=sum zbar^2
    float s0 = (live && cnt > 0.f) ? 1.f : 0.f;
    float s1 = s0 * zbar;
    float s2 = s1 * zbar;
#pragma unroll
    for (int off = 16; off > 0; off >>= 1) {
        s0 += __shfl_down(s0, off);
        s1 += __shfl_down(s1, off);
        s2 += __shfl_down(s2, off);
    }
    if (lane == 0) {
        wred[wave * 3 + 0] = s0;
        wred[wave * 3 + 1] = s1;
        wred[wave * 3 + 2] = s2;
    }
    __syncthreads();
    if (tid == 0) {
        float t0 = 0.f, t1 = 0.f, t2 = 0.f;
        for (int i = 0; i < 8; ++i) {
            t0 += wred[i * 3 + 0];
            t1 += wred[i * 3 + 1];
            t2 += wred[i * 3 + 2];
        }
        partials[(size_t)blockIdx.x * 3 + 0] = t0;
        partials[(size_t)blockIdx.x * 3 + 1] = t1;
        partials[(size_t)blockIdx.x * 3 + 2] = t2;
    }
}

// ---------------------------------------------------------------------------
// Kernel 3: reduce block partials -> zmean, 1/zstd
// ---------------------------------------------------------------------------
__global__ __launch_bounds__(256) void reduce_stats(
    const float* __restrict__ partials, int nblocks, float* __restrict__ stats)
{
    __shared__ float s[3 * 256];
    int tid = threadIdx.x;
    float s0 = 0.f, s1 = 0.f, s2 = 0.f;
    for (int i = tid; i < nblocks; i += 256) {
        s0 += partials[(size_t)i * 3 + 0];
        s1 += partials[(size_t)i * 3 + 1];
        s2 += partials[(size_t)i * 3 + 2];
    }
    s[tid] = s0; s[256 + tid] = s1; s[512 + tid] = s2;
    __syncthreads();
    for (int off = 128; off > 0; off >>= 1) {
        if (tid < off) {
            s[tid]       += s[tid + off];
            s[256 + tid] += s[256 + tid + off];
            s[512 + tid] += s[512 + tid + off];
        }
        __syncthreads();
    }
    if (tid == 0) {
        float S0 = s[0], S1 = s[256], S2 = s[512];
        float nv = (S0 > 0.f) ? S0 : 1.f;          // max(sum valid, 1)
        float zmean = S1 / nv;
        float ssd = S2 - 2.f * zmean * S1 + zmean * zmean * S0;
        ssd = (ssd > 0.f) ? ssd : 0.f;
        float zstd = sqrtf(ssd) + 1e-5f;
        stats[0] = zmean;
        stats[1] = 1.f / zstd;
    }
}

// ---------------------------------------------------------------------------
// Kernel 4: write normalized depth channel
// ---------------------------------------------------------------------------
__global__ void finalize_znorm(const float* __restrict__ zbar,
                               const float* __restrict__ cnt,
                               const float* __restrict__ stats,
                               float* __restrict__ out_feat, int N)
{
    int n = blockIdx.x * blockDim.x + threadIdx.x;
    if (n >= N) return;
    float valid = (cnt[n] > 0.f) ? 1.f : 0.f;
    out_feat[(size_t)n * OUT_CH + 24] = (zbar[n] - stats[0]) * stats[1] * valid;
}

// ---------------------------------------------------------------------------
extern "C" void kernel_launch(void* const* d_in, const int* in_sizes, int n_in,
                              void* d_out, int out_size, void* d_ws, size_t ws_size,
                              hipStream_t stream) {
    const int*   coords = (const int*)d_in[0];     // (N,4) int32
    const float* origin = (const float*)d_in[1];   // (1,3) f32
    const float* feats  = (const float*)d_in[2];   // (V,C,H,W) f32
    const float* proj   = (const float*)d_in[3];   // (V,4,4) f32

    const int N = in_sizes[0] / 4;
    const int V = in_sizes[3] / 16;
    const int ftotal = in_sizes[2];                // V*C*H*W

    float* out_feat = (float*)d_out;               // (N,25)
    float* out_cnt  = out_feat + (size_t)N * OUT_CH;

    // workspace layout
    float* ft       = (float*)d_ws;                        // V*H*W*C
    float* zbar_ws  = ft + (size_t)V * H_IM * W_IM * C_CH; // N
    const int numBlocks = (N + 255) / 256;
    float* partials = zbar_ws + N;                         // numBlocks*3
    float* stats    = partials + (size_t)numBlocks * 3;    // 2

    transpose_feats<<<(ftotal + 255) / 256, 256, 0, stream>>>(feats, ft, ftotal);
    backproject<<<numBlocks, 256, 0, stream>>>(coords, origin, ft, proj,
                                               out_feat, out_cnt, zbar_ws,
                                               partials, N);
    reduce_stats<<<1, 256, 0, stream>>>(partials, numBlocks, stats);
    finalize_znorm<<<(N + 255) / 256, 256, 0, stream>>>(zbar_ws, out_cnt, stats,
                                                        out_feat, N);
}